// LSTM_55748675502949
// MI455X (gfx1250) — compile-verified
//
#include <hip/hip_runtime.h>
#include <hip/hip_bf16.h>

typedef __attribute__((ext_vector_type(2))) float v2f;
typedef __attribute__((ext_vector_type(8))) float v8f;

#define D_IN 8192
#define H    4096
#define COLS 32   // output columns per workgroup (two WMMA M-tiles -> full 128B lines per wave)

// GEMV over rows [kBegin,kEnd) of W (row-major KxH), columns [col0,col0+32).
// A0/A1 = transposed W tiles (A[m][kk] = W[k+kk][col0+(tile*16)+m]),
// B = vector broadcast (B[kk][n] = vec[k+kk] for all n, shared by both tiles).
// D[m][n] = partial out, replicated across n. Full f32 precision.
__device__ __forceinline__ void gemv2_wmma_f32(const float* __restrict__ W,
                                               const float* __restrict__ vec,
                                               int kBegin, int kEnd, int col0,
                                               v8f& acc0, v8f& acc1, int m, int hh) {
  // Make K bounds provably wave-uniform so vec[] loads become scalar loads.
  kBegin = __builtin_amdgcn_readfirstlane(kBegin);
  kEnd   = __builtin_amdgcn_readfirstlane(kEnd);

  // per-lane pointer: row (kBegin + 2*hh), column col0 + m
  const float* wp = W + (size_t)(kBegin + 2 * hh) * H + col0 + m;
#pragma unroll 8
  for (int k = kBegin; k < kEnd; k += 4) {
    // uniform address -> scalar load (merges into s_load_b512 across unroll)
    float4 xs = *reinterpret_cast<const float4*>(vec + k);
    v2f B;                 // broadcast vector: every lane holds its K pair
    B.x = hh ? xs.z : xs.x;
    B.y = hh ? xs.w : xs.y;
    // Weights are streamed exactly once: non-temporal to spare L2 for x/hidden.
    v2f A0, A1;
    A0.x = __builtin_nontemporal_load(wp);           // row k+2hh,   col m
    A0.y = __builtin_nontemporal_load(wp + H);       // row k+2hh+1, col m
    A1.x = __builtin_nontemporal_load(wp + 16);      // row k+2hh,   col 16+m
    A1.y = __builtin_nontemporal_load(wp + H + 16);  // row k+2hh+1, col 16+m
    acc0 = __builtin_amdgcn_wmma_f32_16x16x4_f32(false, A0, false, B,
                                                 (short)0, acc0, false, false);
    acc1 = __builtin_amdgcn_wmma_f32_16x16x4_f32(false, A1, false, B,
                                                 (short)0, acc1, false, false);
    wp += 4 * (size_t)H;
  }
}

__device__ __forceinline__ float sigmoidf_(float z) {
  return 1.0f / (1.0f + expf(-z));
}

__global__ __launch_bounds__(512) void lstm_cell_kernel(
    const float* __restrict__ x, const float* __restrict__ state,
    const float* __restrict__ hidden,
    const float* __restrict__ Wxf, const float* __restrict__ bf, const float* __restrict__ Whf,
    const float* __restrict__ Wxi, const float* __restrict__ bi, const float* __restrict__ Whi,
    const float* __restrict__ Wxc, const float* __restrict__ bc, const float* __restrict__ Whc,
    const float* __restrict__ Wxo, const float* __restrict__ bo, const float* __restrict__ Who,
    float* __restrict__ out) {
  __shared__ float red[16][COLS];  // per-wave partial rows (2 KB)

  const int tid  = threadIdx.x;
  const int wave = tid >> 5;       // 0..15
  const int lane = tid & 31;
  const int m    = lane & 15;      // row of A tile = output column within tile
  const int hh   = lane >> 4;      // K-half within A/B fragment
  const int gate = wave >> 2;      // 0=f 1=i 2=c 3=o
  const int q    = wave & 3;       // K quarter
  const int col0 = blockIdx.x * COLS;

  const float* Wxg = (gate == 0) ? Wxf : (gate == 1) ? Wxi : (gate == 2) ? Wxc : Wxo;
  const float* Whg = (gate == 0) ? Whf : (gate == 1) ? Whi : (gate == 2) ? Whc : Who;

  v8f acc0 = {};
  v8f acc1 = {};
  // x path: D_IN rows split in quarters of 2048
  gemv2_wmma_f32(Wxg, x,      q * (D_IN / 4), (q + 1) * (D_IN / 4), col0, acc0, acc1, m, hh);
  // hidden path: H rows split in quarters of 1024
  gemv2_wmma_f32(Whg, hidden, q * (H / 4),    (q + 1) * (H / 4),    col0, acc0, acc1, m, hh);

  // D layout: VGPR e, lanes 0-15 -> M=e, lanes 16-31 -> M=e+8; value constant over N.
  // lane 0: tile cols 0..7, lane 16: tile cols 8..15 (per tile).
  if (m == 0) {
    float4* d0 = reinterpret_cast<float4*>(&red[wave][hh * 8]);
    d0[0] = make_float4(acc0[0], acc0[1], acc0[2], acc0[3]);
    d0[1] = make_float4(acc0[4], acc0[5], acc0[6], acc0[7]);
    float4* d1 = reinterpret_cast<float4*>(&red[wave][16 + hh * 8]);
    d1[0] = make_float4(acc1[0], acc1[1], acc1[2], acc1[3]);
    d1[1] = make_float4(acc1[4], acc1[5], acc1[6], acc1[7]);
  }
  __syncthreads();

  if (tid < COLS) {
    const int j = col0 + tid;
    float pf = red[0][tid]  + red[1][tid]  + red[2][tid]  + red[3][tid]  + bf[j];
    float pi = red[4][tid]  + red[5][tid]  + red[6][tid]  + red[7][tid]  + bi[j];
    float pc = red[8][tid]  + red[9][tid]  + red[10][tid] + red[11][tid] + bc[j];
    float po = red[12][tid] + red[13][tid] + red[14][tid] + red[15][tid] + bo[j];

    float f = sigmoidf_(pf);
    float i = sigmoidf_(pi);
    float c = tanhf(pc);
    float o = sigmoidf_(po);
    float ns = state[j] * f + c * i;
    out[j] = tanhf(ns) * o;
  }
}

extern "C" void kernel_launch(void* const* d_in, const int* in_sizes, int n_in,
                              void* d_out, int out_size, void* d_ws, size_t ws_size,
                              hipStream_t stream) {
  const float* x      = (const float*)d_in[0];
  const float* state  = (const float*)d_in[1];
  const float* hidden = (const float*)d_in[2];
  const float* Wxf = (const float*)d_in[3];
  const float* bf  = (const float*)d_in[4];
  const float* Whf = (const float*)d_in[5];
  const float* Wxi = (const float*)d_in[6];
  const float* bi  = (const float*)d_in[7];
  const float* Whi = (const float*)d_in[8];
  const float* Wxc = (const float*)d_in[9];
  const float* bc  = (const float*)d_in[10];
  const float* Whc = (const float*)d_in[11];
  const float* Wxo = (const float*)d_in[12];
  const float* bo  = (const float*)d_in[13];
  const float* Who = (const float*)d_in[14];
  float* out = (float*)d_out;

  dim3 grid(H / COLS);   // 128 workgroups
  dim3 block(512);       // 16 waves: 4 gates x 4 K-quarters
  lstm_cell_kernel<<<grid, block, 0, stream>>>(x, state, hidden,
                                               Wxf, bf, Whf,
                                               Wxi, bi, Whi,
                                               Wxc, bc, Whc,
                                               Wxo, bo, Who,
                                               out);
}